// ImportantCategoryLoss_68118181315307
// MI455X (gfx1250) — compile-verified
//
#include <hip/hip_runtime.h>

// ImportantCategoryLoss for MI455X (gfx1250, wave32).
// Bandwidth-bound streaming argmax + gather + deterministic reduction.
//   d_in[0]: log_probs  f32 [65536*1000]
//   d_in[1]: target     i32 [65536]
//   d_in[2]: important_mask bool(u8) [1000]
//   d_out:   f32 scalar = -mean(adjusted)

typedef float v4f __attribute__((ext_vector_type(4)));

#define BATCH_N   65536
#define NCLS      1000
#define NV4       250           // float4 chunks per row (1000/4), row stride 4000B is 16B-aligned
#define WPB       8             // waves per block (256 threads)
#define NBLOCKS   (BATCH_N / WPB) // 8192

__global__ __launch_bounds__(256) void icl_rows_kernel(
    const float* __restrict__ lp,
    const int* __restrict__ target,
    const unsigned char* __restrict__ mask,
    float* __restrict__ partial)
{
    __shared__ float waveSum[WPB];

    const int lane = threadIdx.x & 31;
    const int wave = threadIdx.x >> 5;
    const int row  = blockIdx.x * WPB + wave;

    const float* rp = lp + (size_t)row * NCLS;
    const v4f*   rp4 = (const v4f*)rp;

    float best = -__builtin_inff();
    int   bidx = 0;

    // 250 float4s per row, 32 lanes: lane + 32k, k = 0..7 (lanes >= 26 skip k=7).
    // Coalesced 512B per step; NT hint: 262MB stream > 192MB L2, don't thrash it.
#pragma unroll
    for (int k = 0; k < 8; ++k) {
        const int j = lane + (k << 5);
        if (j < NV4) {
            v4f v = __builtin_nontemporal_load(rp4 + j);
            const int base = j << 2;
            // strict '>' keeps the lowest index within a lane (indices increase with k)
            if (v.x > best) { best = v.x; bidx = base;     }
            if (v.y > best) { best = v.y; bidx = base + 1; }
            if (v.z > best) { best = v.z; bidx = base + 2; }
            if (v.w > best) { best = v.w; bidx = base + 3; }
        }
    }

    // wave32 (val, idx) argmax reduction; tie-break to lowest index (jnp.argmax semantics)
#pragma unroll
    for (int off = 16; off > 0; off >>= 1) {
        const float ov = __shfl_xor(best, off, 32);
        const int   oi = __shfl_xor(bidx, off, 32);
        if (ov > best || (ov == best && oi < bidx)) { best = ov; bidx = oi; }
    }

    if (lane == 0) {
        const int  tgt  = target[row];
        const bool keep = (mask[bidx] != 0) || (mask[tgt] != 0);
        waveSum[wave] = keep ? rp[tgt] : 0.0f;
    }
    __syncthreads();

    if (threadIdx.x == 0) {
        float s = 0.0f;
#pragma unroll
        for (int w = 0; w < WPB; ++w) s += waveSum[w];
        partial[blockIdx.x] = s;   // fixed order -> deterministic
    }
}

__global__ __launch_bounds__(1024) void icl_reduce_kernel(
    const float* __restrict__ partial,
    float* __restrict__ out)
{
    __shared__ float sm[1024];
    float s = 0.0f;
    // fixed strided accumulation order -> deterministic
    for (int i = threadIdx.x; i < NBLOCKS; i += 1024) s += partial[i];
    sm[threadIdx.x] = s;
    __syncthreads();
#pragma unroll
    for (int off = 512; off > 0; off >>= 1) {
        if (threadIdx.x < (unsigned)off) sm[threadIdx.x] += sm[threadIdx.x + off];
        __syncthreads();
    }
    if (threadIdx.x == 0) out[0] = -sm[0] * (1.0f / (float)BATCH_N);
}

extern "C" void kernel_launch(void* const* d_in, const int* in_sizes, int n_in,
                              void* d_out, int out_size, void* d_ws, size_t ws_size,
                              hipStream_t stream) {
    (void)in_sizes; (void)n_in; (void)out_size; (void)ws_size;
    const float*         lp     = (const float*)d_in[0];
    const int*           target = (const int*)d_in[1];
    const unsigned char* mask   = (const unsigned char*)d_in[2];
    float*               out    = (float*)d_out;
    float*               part   = (float*)d_ws;  // 8192 floats = 32KB scratch

    icl_rows_kernel<<<NBLOCKS, 256, 0, stream>>>(lp, target, mask, part);
    icl_reduce_kernel<<<1, 1024, 0, stream>>>(part, out);
}